// freup_AreadinterpolationV2_82824149336372
// MI455X (gfx1250) — compile-verified
//
#include <hip/hip_runtime.h>
#include <hip/hip_bf16.h>

// ---------------------------------------------------------------------------
// freup_AreadinterpolationV2, B=4 C=128 H=W=128, all f32.
// Simplified exactly:  out = wpost * upsample2x(|IFFT2_128(G)|) + bpost,
// where G = Mag*exp(i*Pha), Mag/Pha = 2-layer 1x1-conv MLPs of |F|,angle(F),
// F = FFT2_128(x).  (The 256-point IFFT / d_map / crop collapse analytically.)
//
// Workspace layout (needs 64 MiB):
//   magbuf : 8388608 f32   (mag -> Re(G) -> q, reused in place)
//   phabuf : 8388608 f32   (pha -> Im(G))
// ---------------------------------------------------------------------------

typedef __attribute__((ext_vector_type(2))) float v2f;
typedef __attribute__((ext_vector_type(8))) float v8f;

__device__ __forceinline__ int brev7(int x) { return (int)(__brev((unsigned)x) >> 25); }

// ---------------------------------------------------------------------------
// In-LDS 2D FFT of a 128x128 complex plane.  256 threads.
// Row pass: radix-2 DIT along columns (input pre-bit-reversed in column idx).
// Col pass: radix-2 DIF along rows (natural input, bit-reversed row output).
// sign = -1 forward, +1 inverse (scaling applied by caller).
// ---------------------------------------------------------------------------
__device__ __forceinline__ void fft2d_128(float* re, float* im, int tid, float sign)
{
    const float PI = 3.14159265358979323846f;
    for (int s = 0; s < 7; ++s) {                       // row pass (DIT)
        const int half = 1 << s;
        for (int t = tid; t < 8192; t += 256) {
            const int j   = t & 63;                     // butterfly in row
            const int r   = t >> 6;                     // row
            const int blk = j >> s;
            const int pos = j & (half - 1);
            const int i0  = (r << 7) + blk * (half << 1) + pos;
            const int i1  = i0 + half;
            float sv, cv;
            __sincosf(sign * PI * (float)pos / (float)half, &sv, &cv);
            const float ur = re[i0], ui = im[i0];
            const float vr = re[i1], vi = im[i1];
            const float tr = vr * cv - vi * sv;
            const float ti = vr * sv + vi * cv;
            re[i0] = ur + tr; im[i0] = ui + ti;
            re[i1] = ur - tr; im[i1] = ui - ti;
        }
        __syncthreads();
    }
    for (int s = 6; s >= 0; --s) {                      // column pass (DIF)
        const int half = 1 << s;
        for (int t = tid; t < 8192; t += 256) {
            const int c   = t & 127;                    // column fastest -> bank spread
            const int j   = t >> 7;                     // butterfly among rows
            const int blk = j >> s;
            const int pos = j & (half - 1);
            const int r0  = blk * (half << 1) + pos;
            const int i0  = (r0 << 7) + c;
            const int i1  = i0 + (half << 7);
            float sv, cv;
            __sincosf(sign * PI * (float)pos / (float)half, &sv, &cv);
            const float ur = re[i0], ui = im[i0];
            const float vr = re[i1], vi = im[i1];
            re[i0] = ur + vr; im[i0] = ui + vi;
            const float dr = ur - vr, di = ui - vi;
            re[i1] = dr * cv - di * sv;
            im[i1] = dr * sv + di * cv;
        }
        __syncthreads();
    }
}

// ------------------------- Kernel A: FFT2 -> mag, pha ----------------------
__global__ void __launch_bounds__(256)
fft_mag_pha_kernel(const float* __restrict__ x, float* __restrict__ mag,
                   float* __restrict__ pha)
{
    __shared__ float re[16384];
    __shared__ float im[16384];
    const int tid = threadIdx.x;
    const size_t pofs = (size_t)blockIdx.x << 14;       // plane offset
    const float* xp = x + pofs;
    for (int i = tid; i < 16384; i += 256) {
        const int idx = (i & ~127) | brev7(i & 127);    // bit-reverse column
        re[idx] = xp[i];
        im[idx] = 0.0f;
    }
    __syncthreads();
    fft2d_128(re, im, tid, -1.0f);
    float* mp = mag + pofs;
    float* pp = pha + pofs;
    for (int i = tid; i < 16384; i += 256) {
        const int idx = (brev7(i >> 7) << 7) | (i & 127);  // undo row bit-rev
        const float a = re[idx], b = im[idx];
        mp[i] = sqrtf(a * a + b * b);
        pp[i] = atan2f(b, a);
    }
}

// ------------------------- Kernel C: IFFT2 -> |.|/N ------------------------
__global__ void __launch_bounds__(256)
ifft_abs_kernel(const float* gre, const float* gim, float* q)
{
    __shared__ float re[16384];
    __shared__ float im[16384];
    const int tid = threadIdx.x;
    const size_t pofs = (size_t)blockIdx.x << 14;
    const float* rp = gre + pofs;
    const float* ip = gim + pofs;
    for (int i = tid; i < 16384; i += 256) {
        const int idx = (i & ~127) | brev7(i & 127);
        re[idx] = rp[i];
        im[idx] = ip[i];
    }
    __syncthreads();
    fft2d_128(re, im, tid, +1.0f);
    float* qp = q + pofs;                               // may alias gre: all reads done
    for (int i = tid; i < 16384; i += 256) {
        const int idx = (brev7(i >> 7) << 7) | (i & 127);
        const float a = re[idx], b = im[idx];
        qp[i] = sqrtf(a * a + b * b) * (1.0f / 16384.0f);
    }
}

// ---------------------------------------------------------------------------
// WMMA f32 16x16x4 GEMM helpers.  D[o][n] = sum_c W[o][c] * in[c][n].
// A operand (ISA 7.12.2, 32-bit 16x4): lane m=lane&15, K pair = 2*(lane>>4).
// B operand (4x16): lane n=lane&15, K pair = 2*(lane>>4) across 2 VGPRs.
// C/D: VGPR k holds (M = k + 8*(lane>>4), N = lane&15).
// Weights in LDS with row stride 132 -> conflict-free ds_load_b64 A fetches.
// ---------------------------------------------------------------------------
__device__ __forceinline__ void zero_acc(v8f acc[8])
{
    const v8f z = {0.f, 0.f, 0.f, 0.f, 0.f, 0.f, 0.f, 0.f};
#pragma unroll
    for (int m = 0; m < 8; ++m) acc[m] = z;
}

__device__ __forceinline__ void load_wb(const float* __restrict__ w,
                                        const float* __restrict__ bias,
                                        float* wsh, float* bsh, int tid, int nthr)
{
    for (int i = tid; i < 128 * 128; i += nthr)
        wsh[(i >> 7) * 132 + (i & 127)] = w[i];
    for (int i = tid; i < 128; i += nthr)
        bsh[i] = bias[i];
}

// B operand streamed from global plane [128][16384]; 8 M-tiles reuse it.
__device__ __forceinline__ void wmma_gemm_globalB(const float* __restrict__ wsh,
                                                  const float* inplane, int pix,
                                                  int lane, v8f acc[8])
{
    const int m16  = lane & 15;
    const int koff = (lane >> 4) << 1;
    const float* col = inplane + pix + m16;             // n == lane&15
#pragma unroll 4
    for (int k0 = 0; k0 < 128; k0 += 4) {
        const int kk = k0 + koff;
        v2f bv;
        bv.x = col[(size_t)kk << 14];
        bv.y = col[(size_t)(kk + 1) << 14];
#pragma unroll
        for (int m = 0; m < 8; ++m) {
            const float* wrow = wsh + (m * 16 + m16) * 132 + kk;
            v2f av; av.x = wrow[0]; av.y = wrow[1];
            acc[m] = __builtin_amdgcn_wmma_f32_16x16x4_f32(
                false, av, false, bv, (short)0, acc[m], false, false);
        }
    }
}

// B operand from LDS buffer [128][66] (padded stride 66).
__device__ __forceinline__ void wmma_gemm_ldsB(const float* __restrict__ wsh,
                                               const float* __restrict__ t1,
                                               int ncol, int lane, v8f acc[8])
{
    const int m16  = lane & 15;
    const int koff = (lane >> 4) << 1;
    const float* col = t1 + ncol + m16;
#pragma unroll 4
    for (int k0 = 0; k0 < 128; k0 += 4) {
        const int kk = k0 + koff;
        v2f bv;
        bv.x = col[kk * 66];
        bv.y = col[(kk + 1) * 66];
#pragma unroll
        for (int m = 0; m < 8; ++m) {
            const float* wrow = wsh + (m * 16 + m16) * 132 + kk;
            v2f av; av.x = wrow[0]; av.y = wrow[1];
            acc[m] = __builtin_amdgcn_wmma_f32_16x16x4_f32(
                false, av, false, bv, (short)0, acc[m], false, false);
        }
    }
}

// ------------- Kernel B: both MLPs + complex reconstruction (in place) -----
// 128 threads (4 waves), each wave owns 16 pixels; 1024 blocks.
__global__ void __launch_bounds__(128)
mlp_combine_kernel(const float* __restrict__ w1a, const float* __restrict__ b1a,
                   const float* __restrict__ w2a, const float* __restrict__ b2a,
                   const float* __restrict__ w1p, const float* __restrict__ b1p,
                   const float* __restrict__ w2p, const float* __restrict__ b2p,
                   float* magbuf, float* phabuf)
{
    __shared__ float wsh[128 * 132];
    __shared__ float bsh[128];
    __shared__ float t1[128 * 66];      // layer-1 activations [chan][64 local pix]
    __shared__ float magsh[128 * 66];   // Mag results held for the combine

    const int tid  = threadIdx.x;
    const int wave = tid >> 5;
    const int lane = tid & 31;
    const int b     = blockIdx.x >> 8;                  // 256 blocks / batch
    const int wgpix = (blockIdx.x & 255) << 6;          // 64 pixels / block
    const int pb    = wgpix + (wave << 4);              // wave's pixel base
    const int nloc  = wave << 4;                        // local column base
    const int n     = lane & 15;
    const int hi    = lane >> 4;

    float* magplane = magbuf + ((size_t)b << 21);       // b*128*16384
    float* phaplane = phabuf + ((size_t)b << 21);

    v8f acc[8];

    // ---- amplitude path: t1 = lrelu(W1a*mag + b1a); Mag = W2a*t1 + b2a ----
    load_wb(w1a, b1a, wsh, bsh, tid, 128);
    __syncthreads();
    zero_acc(acc);
    wmma_gemm_globalB(wsh, magplane, pb, lane, acc);
#pragma unroll
    for (int m = 0; m < 8; ++m)
#pragma unroll
        for (int k = 0; k < 8; ++k) {
            const int ch = m * 16 + k + hi * 8;
            float v = acc[m][k] + bsh[ch];
            v = (v > 0.f) ? v : 0.1f * v;
            t1[ch * 66 + nloc + n] = v;
        }
    __syncthreads();
    load_wb(w2a, b2a, wsh, bsh, tid, 128);
    __syncthreads();
    zero_acc(acc);
    wmma_gemm_ldsB(wsh, t1, nloc, lane, acc);
#pragma unroll
    for (int m = 0; m < 8; ++m)
#pragma unroll
        for (int k = 0; k < 8; ++k) {
            const int ch = m * 16 + k + hi * 8;
            magsh[ch * 66 + nloc + n] = acc[m][k] + bsh[ch];
        }
    __syncthreads();

    // ---- phase path, then G = Mag*exp(i*Pha) written over mag/pha ----
    load_wb(w1p, b1p, wsh, bsh, tid, 128);
    __syncthreads();
    zero_acc(acc);
    wmma_gemm_globalB(wsh, phaplane, pb, lane, acc);
#pragma unroll
    for (int m = 0; m < 8; ++m)
#pragma unroll
        for (int k = 0; k < 8; ++k) {
            const int ch = m * 16 + k + hi * 8;
            float v = acc[m][k] + bsh[ch];
            v = (v > 0.f) ? v : 0.1f * v;
            t1[ch * 66 + nloc + n] = v;
        }
    __syncthreads();
    load_wb(w2p, b2p, wsh, bsh, tid, 128);
    __syncthreads();
    zero_acc(acc);
    wmma_gemm_ldsB(wsh, t1, nloc, lane, acc);
#pragma unroll
    for (int m = 0; m < 8; ++m)
#pragma unroll
        for (int k = 0; k < 8; ++k) {
            const int ch = m * 16 + k + hi * 8;
            const float p  = acc[m][k] + bsh[ch];
            const float mg = magsh[ch * 66 + nloc + n];
            float sp, cp;
            __sincosf(p, &sp, &cp);
            const size_t g = ((size_t)ch << 14) + (size_t)(pb + n);
            magplane[g] = mg * cp;                      // Re(G)
            phaplane[g] = mg * sp;                      // Im(G)
        }
}

// -------- Kernel D: out = wpost * q + bpost, nearest-2x replicated stores --
__global__ void __launch_bounds__(256)
post_upsample_kernel(const float* __restrict__ wpost, const float* __restrict__ bpost,
                     const float* __restrict__ q, float* __restrict__ out)
{
    __shared__ float wsh[128 * 132];
    __shared__ float bsh[128];
    const int tid  = threadIdx.x;
    const int wave = tid >> 5;
    const int lane = tid & 31;
    const int b     = blockIdx.x >> 7;                  // 128 blocks / batch
    const int wgpix = (blockIdx.x & 127) << 7;          // 128 pixels / block
    const int pb    = wgpix + (wave << 4);
    const int n     = lane & 15;
    const int hi    = lane >> 4;

    load_wb(wpost, bpost, wsh, bsh, tid, 256);
    __syncthreads();

    v8f acc[8];
    zero_acc(acc);
    wmma_gemm_globalB(wsh, q + ((size_t)b << 21), pb, lane, acc);

    const int pix = pb + n;
    const int px  = pix >> 7;
    const int py  = pix & 127;
#pragma unroll
    for (int m = 0; m < 8; ++m)
#pragma unroll
        for (int k = 0; k < 8; ++k) {
            const int ch = m * 16 + k + hi * 8;
            const float v = acc[m][k] + bsh[ch];
            const float2 vv = make_float2(v, v);
            float* o = out + (size_t)(b * 128 + ch) * 65536
                           + (size_t)(px << 1) * 256 + (py << 1);
            *(float2*)o         = vv;                   // row 2*px
            *(float2*)(o + 256) = vv;                   // row 2*px+1
        }
}

// ---------------------------------------------------------------------------
extern "C" void kernel_launch(void* const* d_in, const int* in_sizes, int n_in,
                              void* d_out, int out_size, void* d_ws, size_t ws_size,
                              hipStream_t stream)
{
    (void)in_sizes; (void)n_in; (void)out_size; (void)ws_size;
    const float* x     = (const float*)d_in[0];
    const float* w1a   = (const float*)d_in[1];
    const float* b1a   = (const float*)d_in[2];
    const float* w2a   = (const float*)d_in[3];
    const float* b2a   = (const float*)d_in[4];
    const float* w1p   = (const float*)d_in[5];
    const float* b1p   = (const float*)d_in[6];
    const float* w2p   = (const float*)d_in[7];
    const float* b2p   = (const float*)d_in[8];
    const float* wpost = (const float*)d_in[9];
    const float* bpost = (const float*)d_in[10];
    float* out = (float*)d_out;

    float* magbuf = (float*)d_ws;                       // 32 MiB
    float* phabuf = magbuf + 8388608;                   // 32 MiB  (ws >= 64 MiB)

    fft_mag_pha_kernel<<<512, 256, 0, stream>>>(x, magbuf, phabuf);
    mlp_combine_kernel<<<1024, 128, 0, stream>>>(w1a, b1a, w2a, b2a,
                                                 w1p, b1p, w2p, b2p,
                                                 magbuf, phabuf);
    ifft_abs_kernel<<<512, 256, 0, stream>>>(magbuf, phabuf, magbuf);
    post_upsample_kernel<<<512, 256, 0, stream>>>(wpost, bpost, magbuf, out);
}